// TempFuseDilateAttention_1511828488302
// MI455X (gfx1250) — compile-verified
//
#include <hip/hip_runtime.h>
#include <hip/hip_bf16.h>

typedef float v2f __attribute__((ext_vector_type(2)));
typedef float v8f __attribute__((ext_vector_type(8)));

#define HW   40000      // 200*200
#define CDIM 128
#define SCALE_F 0.17677669529663687f   // 32^-0.5

// ---------------------------------------------------------------------------
// Kernel A: half of the 1x1 conv (o in [ohalf*128, ohalf*128+128)).
// Output layout kvh[n][o_local][hw]  (n = bl*2 + ti, 4 planes of 128*40000).
// GEMM per n: D[o][hw] = sum_c W[o][c] * X[hw][c], via v_wmma_f32_16x16x4_f32.
// Block: 256 threads = 8 waves; block tile = 32 o x 64 hw; wave tile 16x16.
// ---------------------------------------------------------------------------
__global__ __launch_bounds__(256)
void conv_half_kernel(const float* __restrict__ x,    // [4][40000][128]
                      const float* __restrict__ wkv,  // [256][128]
                      float* __restrict__ kvh,        // [4][128][40000]
                      int ohalf)
{
    __shared__ __align__(16) float Wt[32 * 132];
    __shared__ __align__(16) float Xt[64 * 132];

    const int n   = blockIdx.z;
    const int o0  = blockIdx.y * 32;        // local o within this half
    const int hw0 = blockIdx.x * 64;        // 625 * 64 == 40000 exactly
    const int tid = threadIdx.x;

    // stage W rows [ohalf*128+o0 .. +32) x 128  (32 rows * 32 float4)
    {
        const float* src = wkv + (size_t)(ohalf * 128 + o0) * CDIM;
        for (int idx = tid; idx < 32 * 32; idx += 256) {
            int r = idx >> 5, c4 = (idx & 31) << 2;
            *(float4*)&Wt[r * 132 + c4] = *(const float4*)(src + r * CDIM + c4);
        }
    }
    // stage X rows [hw0 .. hw0+64) x 128
    {
        const float* src = x + ((size_t)n * HW + hw0) * CDIM;
        for (int idx = tid; idx < 64 * 32; idx += 256) {
            int r = idx >> 5, c4 = (idx & 31) << 2;
            *(float4*)&Xt[r * 132 + c4] = *(const float4*)(src + (size_t)r * CDIM + c4);
        }
    }
    __syncthreads();

    const int wv = tid >> 5;           // wave 0..7
    const int lane = tid & 31;
    const int lo = lane & 15, hi = lane >> 4;
    const int ot = wv >> 2;            // o tile 0..1
    const int hs = wv & 3;             // hw subtile 0..3

    const float* Arow = &Wt[(ot * 16 + lo) * 132];   // A[M=lo][K], M = o
    const float* Brow = &Xt[(hs * 16 + lo) * 132];   // B[K][N=lo], N = hw (X^T)
    v8f acc = {0.f, 0.f, 0.f, 0.f, 0.f, 0.f, 0.f, 0.f};

#if __has_builtin(__builtin_amdgcn_wmma_f32_16x16x4_f32)
#pragma unroll
    for (int k0 = 0; k0 < 128; k0 += 4) {
        v2f a = *(const v2f*)(Arow + k0 + 2 * hi);   // K = 2*hi + {0,1}
        v2f b = *(const v2f*)(Brow + k0 + 2 * hi);
        acc = __builtin_amdgcn_wmma_f32_16x16x4_f32(
            false, a, false, b, (short)0, acc, false, false);
    }
#else
    // scalar fallback (keeps round alive if builtin name is wrong)
#pragma unroll
    for (int g = 0; g < 8; ++g) {
        float s = 0.f;
        const float* ar = &Wt[(ot * 16 + g + 8 * hi) * 132];
        const float* br = &Xt[(hs * 16 + lo) * 132];
        for (int k = 0; k < 128; ++k) s += ar[k] * br[k];
        acc[g] = s;
    }
#endif

    // D[M=g+8*hi][N=lo] -> kvh[n][o0 + ot*16 + g + 8*hi][hw0 + hs*16 + lo]
    float* dst = kvh + ((size_t)n * 128 + (o0 + ot * 16 + 8 * hi)) * HW
                     + (hw0 + hs * 16 + lo);
#pragma unroll
    for (int g = 0; g < 8; ++g) dst[(size_t)g * HW] = acc[g];
}

// ---------------------------------------------------------------------------
// Kernel B: one dilate-attention pass (faithful scrambled indexing).
// One thread per (b, n, p). Online softmax over the 9 taps; gather offsets
// computed once and reused for K and V.
// ---------------------------------------------------------------------------
__global__ __launch_bounds__(256)
void attn_pass_kernel(const float* __restrict__ qin,  // [2][40000*128] flat
                      const float* __restrict__ kvh,  // [4][128][40000]
                      float* __restrict__ qout,       // [2][40000][128]
                      int dil)
{
    const int p = blockIdx.x * 256 + threadIdx.x;
    if (p >= HW) return;
    const int n = blockIdx.y;   // head 0..3
    const int b = blockIdx.z;   // batch 0..1

    const float* qb = qin + (size_t)b * (HW * CDIM);
    const float* Kb = kvh + (size_t)(b * 2 + 0) * (128 * HW);  // frame 0 half-block
    const float* Vb = kvh + (size_t)(b * 2 + 1) * (128 * HW);  // frame 1 half-block

    float qv[32];
#pragma unroll
    for (int dd = 0; dd < 32; ++dd)
        qv[dd] = qb[(size_t)(n * 32 + dd) * HW + p];   // channel-major read of q

    float m = -1e30f, denom = 0.f;
    float acc[32];
#pragma unroll
    for (int dd = 0; dd < 32; ++dd) acc[dd] = 0.f;

    for (int kk2 = 0; kk2 < 9; ++kk2) {
        int offs[32];
        float s = 0.f;
#pragma unroll
        for (int dd = 0; dd < 32; ++dd) {
            const int cl = n * 32 + dd;                    // 0..127
            const int T  = (cl * 9 + kk2) * HW + p;        // < 2^31
            const int chunk = T / 25600;                   // c'*9 + kk, < 1800
            const int sp    = T - chunk * 25600;           // < 25600
            const int cp = chunk / 9;                      // actual-H index
            const int kk = chunk - cp * 9;
            const int ki = kk / 3, kj = kk - ki * 3;
            const int w2 = (sp >> 7) + (ki - 1) * dil;     // actual-W index (padded)
            const int c3 = (sp & 127) + (kj - 1) * dil;    // actual-C index (padded)
            int off = -1;
            if ((unsigned)w2 < 200u && (unsigned)c3 < 128u)
                off = cp * 25600 + (w2 << 7) + c3;
            offs[dd] = off;
            const float kval = (off >= 0) ? Kb[off] : 0.f;
            s += qv[dd] * kval;
        }
        s *= SCALE_F;
        const float mn = fmaxf(m, s);
        const float cs = __expf(m - mn);
        const float w  = __expf(s - mn);
        denom = denom * cs + w;
#pragma unroll
        for (int dd = 0; dd < 32; ++dd) {
            const float vval = (offs[dd] >= 0) ? Vb[offs[dd]] : 0.f;
            acc[dd] = acc[dd] * cs + w * vval;
        }
        m = mn;
    }

    const float inv = 1.f / denom;
    float* ob = qout + (size_t)b * (HW * CDIM) + (size_t)p * CDIM + n * 32;
#pragma unroll
    for (int g4 = 0; g4 < 8; ++g4) {
        float4 v;
        v.x = acc[g4 * 4 + 0] * inv;
        v.y = acc[g4 * 4 + 1] * inv;
        v.z = acc[g4 * 4 + 2] * inv;
        v.w = acc[g4 * 4 + 3] * inv;
        *(float4*)(ob + g4 * 4) = v;
    }
}

// ---------------------------------------------------------------------------
// Kernel C: output projection GEMM: out[r][co] = sum_c q[r][c]*P[co][c] + bias
// Same WMMA tiling as kernel A (A = q rows, B = P rows as K x N).
// ---------------------------------------------------------------------------
__global__ __launch_bounds__(256)
void proj_kernel(const float* __restrict__ qf,   // [80000][128]
                 const float* __restrict__ pw,   // [128][128]
                 const float* __restrict__ pb,   // [128]
                 float* __restrict__ out)        // [80000][128]
{
    __shared__ __align__(16) float Pt[32 * 132];
    __shared__ __align__(16) float Qt[64 * 132];

    const int r0  = blockIdx.x * 64;     // 1250 * 64 == 80000 exactly
    const int co0 = blockIdx.y * 32;
    const int tid = threadIdx.x;

    for (int idx = tid; idx < 32 * 32; idx += 256) {
        int r = idx >> 5, c4 = (idx & 31) << 2;
        *(float4*)&Pt[r * 132 + c4] = *(const float4*)(pw + (size_t)(co0 + r) * CDIM + c4);
    }
    for (int idx = tid; idx < 64 * 32; idx += 256) {
        int r = idx >> 5, c4 = (idx & 31) << 2;
        *(float4*)&Qt[r * 132 + c4] = *(const float4*)(qf + (size_t)(r0 + r) * CDIM + c4);
    }
    __syncthreads();

    const int wv = tid >> 5, lane = tid & 31;
    const int lo = lane & 15, hi = lane >> 4;
    const int ct = wv >> 2;   // co tile 0..1
    const int rs = wv & 3;    // row subtile 0..3

    const float* Arow = &Qt[(rs * 16 + lo) * 132];
    const float* Brow = &Pt[(ct * 16 + lo) * 132];
    const float bv = pb[co0 + ct * 16 + lo];     // N = lo -> same bias for all g
    v8f acc = {bv, bv, bv, bv, bv, bv, bv, bv};

#if __has_builtin(__builtin_amdgcn_wmma_f32_16x16x4_f32)
#pragma unroll
    for (int k0 = 0; k0 < 128; k0 += 4) {
        v2f a = *(const v2f*)(Arow + k0 + 2 * hi);
        v2f b = *(const v2f*)(Brow + k0 + 2 * hi);
        acc = __builtin_amdgcn_wmma_f32_16x16x4_f32(
            false, a, false, b, (short)0, acc, false, false);
    }
#else
#pragma unroll
    for (int g = 0; g < 8; ++g) {
        float s = bv;
        const float* ar = &Qt[(rs * 16 + g + 8 * hi) * 132];
        const float* br = &Pt[(ct * 16 + lo) * 132];
        for (int k = 0; k < 128; ++k) s += ar[k] * br[k];
        acc[g] = s;
    }
#endif

    float* dst = out + (size_t)(r0 + rs * 16 + 8 * hi) * CDIM + (co0 + ct * 16 + lo);
#pragma unroll
    for (int g = 0; g < 8; ++g) dst[(size_t)g * CDIM] = acc[g];
}

// ---------------------------------------------------------------------------
extern "C" void kernel_launch(void* const* d_in, const int* in_sizes, int n_in,
                              void* d_out, int out_size, void* d_ws, size_t ws_size,
                              hipStream_t stream) {
    const float* q   = (const float*)d_in[0];   // [2][200][200][128]
    const float* x   = (const float*)d_in[1];   // [2][2][200][200][128]
    const float* wkv = (const float*)d_in[2];   // [256][128]
    const float* pw  = (const float*)d_in[3];   // [128][128]
    const float* pb  = (const float*)d_in[4];   // [128]
    float* out = (float*)d_out;                 // [2][200][200][128]

    float* kvh = (float*)d_ws;                              // 4*128*40000 f32 = 81.92 MB
    float* q1  = kvh + (size_t)4 * 128 * HW;                // 10.24M f32 = 40.96 MB
    float* q2  = q1 + (size_t)2 * HW * CDIM;                // 40.96 MB

    dim3 blk(256);
    dim3 cg(625, 4, 4);    // conv: hw chunks of 64, o chunks of 32 (within half), n
    dim3 ag(157, 4, 2);    // attn: p chunks of 256, head n, batch b
    dim3 pg(1250, 4);      // proj: row chunks of 64, co chunks of 32

    // pass 0 (dilation 1): K/V come from conv channels o in [0,128)
    conv_half_kernel<<<cg, blk, 0, stream>>>(x, wkv, kvh, 0);
    attn_pass_kernel<<<ag, blk, 0, stream>>>(q, kvh, q1, 1);
    // pass 1 (dilation 2): K/V come from conv channels o in [128,256)
    conv_half_kernel<<<cg, blk, 0, stream>>>(x, wkv, kvh, 1);
    attn_pass_kernel<<<ag, blk, 0, stream>>>(q1, kvh, q2, 2);
    // final projection
    proj_kernel<<<pg, blk, 0, stream>>>(q2, pw, pb, out);
}